// CachedGCN_19688130085401
// MI455X (gfx1250) — compile-verified
//
#include <hip/hip_runtime.h>

typedef float v2f __attribute__((ext_vector_type(2)));
typedef float v8f __attribute__((ext_vector_type(8)));

constexpr int N_NODES = 100000;
constexpr int N_EDGES = 1600000;
constexpr int IN_DIM  = 64;
constexpr int HIDDEN  = 128;
constexpr int OUT_DIM = 64;

// ---------------------------------------------------------------------------
// SpMM scatter: z[row[e], :] += val[e] * x[col[e], :]
// One wave32 per 2 edges; lane l owns D/32 contiguous floats of the row, so
// the gather of x[col] is one coalesced b64 load per lane, and the scatter is
// VEC global_atomic_add_f32 ops that resolve at the L2 atomic units (entire z
// fits in the 192MB L2). The second edge's feature row is prefetched
// (global_prefetch_b8) while the first is processed.
// ---------------------------------------------------------------------------
template<int D>
__global__ void spmm_scatter(const int* __restrict__ row,
                             const int* __restrict__ col,
                             const float* __restrict__ val,
                             const float* __restrict__ x,
                             float* __restrict__ z,
                             int n_edges) {
  constexpr int VEC = D / 32;                  // floats per lane
  const int wave = threadIdx.x >> 5;
  const int lane = threadIdx.x & 31;
  const long e0 = ((long)blockIdx.x * (blockDim.x >> 5) + wave) * 2;

#pragma unroll
  for (int j = 0; j < 2; ++j) {
    const long e = e0 + j;
    if (e >= n_edges) return;

    if (j == 0 && e + 1 < n_edges) {
      // warm L2/L0 for the next edge's gather row
      const int cnext = col[e + 1];
      __builtin_prefetch(x + (long)cnext * D + lane * VEC, 0, 1);
    }

    const int   r = row[e];
    const int   c = col[e];
    const float v = val[e];

    const float* __restrict__ xs = x + (long)c * D + lane * VEC;
    float* __restrict__       zs = z + (long)r * D + lane * VEC;

    float xv[VEC];
#pragma unroll
    for (int i = 0; i < VEC; ++i) xv[i] = xs[i];   // vectorizes to b64
#pragma unroll
    for (int i = 0; i < VEC; ++i) atomicAdd(zs + i, v * xv[i]);
  }
}

// ---------------------------------------------------------------------------
// Dense GEMM via V_WMMA_F32_16X16X4_F32 (wave32, full fp32 precision).
//   Out[M x N] = act(A[M x K] @ W[K x N])
// Each wave owns one 16-row M-strip and computes ALL N/16 column tiles,
// loading each A fragment once and reusing it across N-tiles (A global
// traffic cut by N/16 vs one-tile-per-wave). W (32KB) staged in LDS once per
// block, PAIR-INTERLEAVED: element (k,n) at lw[(k>>1)*2N + 2n + (k&1)], so a
// B fragment (W[k][n], W[k+1][n]) is one aligned ds_load_b64 straight into an
// even VGPR pair — no repacking movs before the WMMA.
//
// ISA 7.12.2 fragment layouts (wave32):
//   A 16x4 f32 : lanes 0-15 -> M=lane, VGPR0=K0,VGPR1=K1 ; lanes 16-31 -> K2,K3
//   B 4x16 f32 : mirrored  -> lanes 0-15 carry K0/K1 rows, lanes 16-31 K2/K3
//   C/D 16x16  : VGPR r -> M = r + 8*(lane>=16), N = lane&15
// ---------------------------------------------------------------------------
template<int K, int N, bool RELU, int WAVES>
__global__ void gemm_wmma(const float* __restrict__ A,
                          const float* __restrict__ W,
                          float* __restrict__ Out,
                          int mtiles) {
  constexpr int NT = N / 16;                   // N-tiles per wave
  __shared__ float lw[K * N];
  for (int i = threadIdx.x; i < K * N; i += blockDim.x) {
    const int k = i / N;
    const int n = i - k * N;
    lw[(k >> 1) * (2 * N) + 2 * n + (k & 1)] = W[i];
  }
  __syncthreads();

  const int wave  = threadIdx.x >> 5;
  const int lane  = threadIdx.x & 31;
  const int half  = lane >> 4;                 // 0: lanes 0-15, 1: lanes 16-31
  const int l15   = lane & 15;
  const int koff  = half * 2;                  // K sub-offset owned by this half
  const int mtile = blockIdx.x * WAVES + wave;
  if (mtile >= mtiles) return;                 // wave-uniform; EXEC stays all-1s
  const int mbase = mtile * 16;

  const float* __restrict__ arow = A + (long)(mbase + l15) * K;

  v8f acc[NT];
#pragma unroll
  for (int nt = 0; nt < NT; ++nt) acc[nt] = (v8f){};

#pragma unroll
  for (int k0 = 0; k0 < K; k0 += 4) {
    v2f a;
    a.x = arow[k0 + koff];                     // one b64 load, reused NT times
    a.y = arow[k0 + koff + 1];
    // k0+koff is even: (row k0+koff, row k0+koff+1) form one interleaved pair
    const float* __restrict__ bpair = lw + ((k0 + koff) >> 1) * (2 * N);
#pragma unroll
    for (int nt = 0; nt < NT; ++nt) {
      v2f b = *(const v2f*)(bpair + 2 * (nt * 16 + l15));   // ds_load_b64
      acc[nt] = __builtin_amdgcn_wmma_f32_16x16x4_f32(
          /*neg_a=*/false, a, /*neg_b=*/false, b,
          /*c_mod=*/(short)0, acc[nt], /*reuse_a=*/false, /*reuse_b=*/false);
    }
  }

#pragma unroll
  for (int nt = 0; nt < NT; ++nt) {
#pragma unroll
    for (int r = 0; r < 8; ++r) {
      float o = acc[nt][r];
      if (RELU) o = o > 0.f ? o : 0.f;
      Out[(long)(mbase + r + half * 8) * N + nt * 16 + l15] = o;
    }
  }
}

// ---------------------------------------------------------------------------
// Pipeline (uses associativity: (A@h)@W3 == A@(h@W3), halving layer-2 SpMM
// gather bytes and atomics by doing the SpMM at width 64 instead of 128):
//   z1 = A@X          (spmm, D=64)
//   h  = relu(z1@W1)  (WMMA, K=64, N=128)
//   g  = h@W3         (WMMA, K=128, N=64)
//   out= A@g          (spmm, D=64, scattered straight into d_out)
// d_ws layout: z1 [N*64] | h [N*128] | g [N*64]  (= 102.4 MB)
// ---------------------------------------------------------------------------
extern "C" void kernel_launch(void* const* d_in, const int* in_sizes, int n_in,
                              void* d_out, int out_size, void* d_ws, size_t ws_size,
                              hipStream_t stream) {
  const float* features = (const float*)d_in[0];
  const int*   edge_row = (const int*)d_in[1];
  const int*   edge_col = (const int*)d_in[2];
  const float* edge_val = (const float*)d_in[3];
  const float* weight1  = (const float*)d_in[4];
  const float* weight3  = (const float*)d_in[5];
  float* out = (float*)d_out;

  float* z1 = (float*)d_ws;
  float* h  = z1 + (size_t)N_NODES * IN_DIM;
  float* g  = h  + (size_t)N_NODES * HIDDEN;

  hipMemsetAsync(z1,  0, sizeof(float) * (size_t)N_NODES * IN_DIM, stream);
  hipMemsetAsync(out, 0, sizeof(float) * (size_t)out_size,         stream);

  // --- SpMM 1: z1 = A @ X ---
  constexpr int WPB = 8;                       // waves per block
  const long edge_waves  = (N_EDGES + 1) / 2;  // 2 edges per wave
  const int  edge_blocks = (int)((edge_waves + WPB - 1) / WPB);
  spmm_scatter<IN_DIM><<<edge_blocks, WPB * 32, 0, stream>>>(
      edge_row, edge_col, edge_val, features, z1, N_EDGES);

  // --- GEMM 1: h = relu(z1 @ W1) ---
  constexpr int MT = N_NODES / 16;             // 6250 M-tiles (exact)
  constexpr int GW = 8;                        // M-strips (waves) per block
  gemm_wmma<IN_DIM, HIDDEN, true, GW><<<(MT + GW - 1) / GW, GW * 32, 0, stream>>>(
      z1, weight1, h, MT);

  // --- GEMM 2 (reassociated): g = h @ W3 ---
  gemm_wmma<HIDDEN, OUT_DIM, false, GW><<<(MT + GW - 1) / GW, GW * 32, 0, stream>>>(
      h, weight3, g, MT);

  // --- SpMM 2: out = A @ g ---
  spmm_scatter<OUT_DIM><<<edge_blocks, WPB * 32, 0, stream>>>(
      edge_row, edge_col, edge_val, g, out, N_EDGES);
}